// StationFlowGCN3_63015760166990
// MI455X (gfx1250) — compile-verified
//
#include <hip/hip_runtime.h>

typedef float v2f __attribute__((ext_vector_type(2)));
typedef float v8f __attribute__((ext_vector_type(8)));

#define N_NODES 100000
#define N_EDGES 3200000

// ---------------- normalization prep ----------------

__global__ void deg_init_kernel(float* __restrict__ deg, int n) {
  int i = blockIdx.x * blockDim.x + threadIdx.x;
  if (i < n) deg[i] = 1.0f;  // self-loop weight
}

__global__ void deg_edges_kernel(const long long* __restrict__ ei,
                                 const float* __restrict__ ew,
                                 float* __restrict__ deg, int E) {
  int e = blockIdx.x * blockDim.x + threadIdx.x;
  if (e < E) {
    long long d = ei[(long long)E + e];
    atomicAdd(&deg[d], ew[e]);
  }
}

__global__ void dinv_kernel(float* __restrict__ dinv, float* __restrict__ dinv2, int n) {
  int i = blockIdx.x * blockDim.x + threadIdx.x;
  if (i < n) {
    float d = dinv[i];
    float r = (d > 0.0f) ? rsqrtf(d) : 0.0f;
    dinv[i]  = r;
    dinv2[i] = r * r;  // self-loop norm
  }
}

__global__ void norm_kernel(const long long* __restrict__ ei,
                            const float* __restrict__ ew,
                            const float* __restrict__ dinv,
                            float* __restrict__ nrm, int E) {
  int e = blockIdx.x * blockDim.x + threadIdx.x;
  if (e < E) {
    long long s = ei[e];
    long long d = ei[(long long)E + e];
    nrm[e] = dinv[s] * ew[e] * dinv[d];
  }
}

// Zero-pad W [K][FO] -> Wp [K][16]
__global__ void pad_w_kernel(const float* __restrict__ W, float* __restrict__ Wp,
                             int K, int FO) {
  int i = blockIdx.x * blockDim.x + threadIdx.x;
  if (i < K * 16) {
    int k = i >> 4, n = i & 15;
    Wp[i] = (n < FO) ? W[k * FO + n] : 0.0f;
  }
}

// ---------------- WMMA fp32 GEMM: C[M,16] = A[M,K] @ Bp[K,16] ----------------
// One wave per 16-row tile. V_WMMA_F32_16X16X4_F32, K/4 steps, exact fp32.

template <int K, int LDA>
__global__ __launch_bounds__(256) void gemm_wmma_kernel(
    const float* __restrict__ A, const float* __restrict__ Bp,
    float* __restrict__ C, int nrows) {
  static_assert(K % 4 == 0, "K must be multiple of 4");
  const int lane = threadIdx.x & 31;
  const int wave = threadIdx.x >> 5;
  const int tile = blockIdx.x * (blockDim.x >> 5) + wave;
  const int row0 = tile << 4;
  if (row0 >= nrows) return;           // wave-uniform: EXEC all-ones for WMMA
  const int hf = lane >> 4;            // lane-half selects K pair
  const int m  = lane & 15;            // A: row   B: column

  v8f c = {};
#pragma unroll
  for (int k0 = 0; k0 < K; k0 += 4) {
    // A 16x4 fragment: lanes 0-15 -> K = k0,k0+1 ; lanes 16-31 -> K = k0+2,k0+3
    const float* ap = A + (size_t)(row0 + m) * LDA + (k0 + 2 * hf);
    v2f a; a.x = ap[0]; a.y = ap[1];
    // B 4x16 fragment: lane = column, half selects rows (k0,k0+1) vs (k0+2,k0+3)
    const float* bp = Bp + (size_t)(k0 + 2 * hf) * 16 + m;
    v2f b; b.x = bp[0]; b.y = bp[16];
    c = __builtin_amdgcn_wmma_f32_16x16x4_f32(false, a, false, b, (short)0, c,
                                              false, false);
  }
  // D layout: VGPR r holds row (r + 8*half), column = lane&15
  float* cp = C + (size_t)(row0 + 8 * hf) * 16 + m;
#pragma unroll
  for (int r = 0; r < 8; ++r) cp[(size_t)r * 16] = c[r];
}

// ---------------- aggregation ----------------

// agg[i,:] = hw[i,:] * dinv2[i]  (self-loop term), float4 vectorized over N*16
__global__ void agg_init_kernel(const float4* __restrict__ hw,
                                const float* __restrict__ dinv2,
                                float4* __restrict__ agg, int n4) {
  int i = blockIdx.x * blockDim.x + threadIdx.x;
  if (i < n4) {
    float s = dinv2[i >> 2];
    float4 v = hw[i];
    v.x *= s; v.y *= s; v.z *= s; v.w *= s;
    agg[i] = v;
  }
}

template <int FO>
__global__ void scatter_kernel(const long long* __restrict__ ei,
                               const float* __restrict__ nrm,
                               const float* __restrict__ hw,
                               float* __restrict__ agg, int E) {
  int e = blockIdx.x * blockDim.x + threadIdx.x;
  if (e >= E) return;
  long long s = ei[e];
  long long d = ei[(long long)E + e];
  float w = nrm[e];
  const float* hs = hw + (size_t)s * 16;
  float* ad = agg + (size_t)d * 16;
#pragma unroll
  for (int f = 0; f < FO; ++f) atomicAdd(ad + f, hs[f] * w);
}

template <int FO>
__global__ void bias_relu_kernel(const float* __restrict__ agg,
                                 const float* __restrict__ b,
                                 float* __restrict__ h, int n16) {
  int i = blockIdx.x * blockDim.x + threadIdx.x;
  if (i < n16) {
    int col = i & 15;
    float v = 0.0f;
    if (col < FO) v = fmaxf(agg[i] + b[col], 0.0f);
    h[i] = v;  // padded columns stay zero
  }
}

__global__ void final_kernel(const float* __restrict__ agg,
                             const float* __restrict__ b,
                             float* __restrict__ out, int n4) {
  int i = blockIdx.x * blockDim.x + threadIdx.x;
  if (i < n4) {
    int node = i >> 2, col = i & 3;
    out[i] = fmaxf(agg[(size_t)node * 16 + col] + b[col], 0.0f);
  }
}

// ---------------- launcher ----------------

extern "C" void kernel_launch(void* const* d_in, const int* in_sizes, int n_in,
                              void* d_out, int out_size, void* d_ws, size_t ws_size,
                              hipStream_t stream) {
  (void)in_sizes; (void)n_in; (void)out_size; (void)ws_size;
  const float*     x  = (const float*)d_in[0];
  const long long* ei = (const long long*)d_in[1];
  const float*     ew = (const float*)d_in[2];
  const float* W1 = (const float*)d_in[3];  const float* b1 = (const float*)d_in[4];
  const float* W2 = (const float*)d_in[5];  const float* b2 = (const float*)d_in[6];
  const float* W3 = (const float*)d_in[7];  const float* b3 = (const float*)d_in[8];
  const float* W4 = (const float*)d_in[9];  const float* b4 = (const float*)d_in[10];
  float* out = (float*)d_out;

  const int N = N_NODES, E = N_EDGES;

  float* ws = (float*)d_ws;
  size_t off = 0;
  float* nrm   = ws + off; off += (size_t)E;
  float* dinv  = ws + off; off += (size_t)N;      // holds deg, then dinv in-place
  float* dinv2 = ws + off; off += (size_t)N;
  float* hw    = ws + off; off += (size_t)N * 16;
  float* agg   = ws + off; off += (size_t)N * 16;
  float* hbuf  = ws + off; off += (size_t)N * 16;
  float* Wp1   = ws + off; off += 64 * 16;
  float* Wp2   = ws + off; off += 16 * 16;
  float* Wp3   = ws + off; off += 8 * 16;
  float* Wp4   = ws + off; off += 8 * 16;

  const int T = 256;
  const int gN  = (N + T - 1) / T;
  const int gE  = (E + T - 1) / T;
  const int g16 = (N * 16 + T - 1) / T;
  const int g4  = (N * 4 + T - 1) / T;
  const int tiles = (N + 15) / 16;                 // 6250 exact
  const int gG = (tiles + 7) / 8;                  // 8 waves / block

  // normalization
  deg_init_kernel<<<gN, T, 0, stream>>>(dinv, N);
  deg_edges_kernel<<<gE, T, 0, stream>>>(ei, ew, dinv, E);
  dinv_kernel<<<gN, T, 0, stream>>>(dinv, dinv2, N);
  norm_kernel<<<gE, T, 0, stream>>>(ei, ew, dinv, nrm, E);

  // padded weights
  pad_w_kernel<<<(64 * 16 + T - 1) / T, T, 0, stream>>>(W1, Wp1, 64, 16);
  pad_w_kernel<<<1, T, 0, stream>>>(W2, Wp2, 16, 8);
  pad_w_kernel<<<1, T, 0, stream>>>(W3, Wp3, 8, 8);
  pad_w_kernel<<<1, T, 0, stream>>>(W4, Wp4, 8, 4);

  // layer 1: 64 -> 16
  gemm_wmma_kernel<64, 64><<<gG, 256, 0, stream>>>(x, Wp1, hw, N);
  agg_init_kernel<<<g4, T, 0, stream>>>((const float4*)hw, dinv2, (float4*)agg, N * 4);
  scatter_kernel<16><<<gE, T, 0, stream>>>(ei, nrm, hw, agg, E);
  bias_relu_kernel<16><<<g16, T, 0, stream>>>(agg, b1, hbuf, N * 16);

  // layer 2: 16 -> 8
  gemm_wmma_kernel<16, 16><<<gG, 256, 0, stream>>>(hbuf, Wp2, hw, N);
  agg_init_kernel<<<g4, T, 0, stream>>>((const float4*)hw, dinv2, (float4*)agg, N * 4);
  scatter_kernel<8><<<gE, T, 0, stream>>>(ei, nrm, hw, agg, E);
  bias_relu_kernel<8><<<g16, T, 0, stream>>>(agg, b2, hbuf, N * 16);

  // layer 3: 8 -> 8
  gemm_wmma_kernel<8, 16><<<gG, 256, 0, stream>>>(hbuf, Wp3, hw, N);
  agg_init_kernel<<<g4, T, 0, stream>>>((const float4*)hw, dinv2, (float4*)agg, N * 4);
  scatter_kernel<8><<<gE, T, 0, stream>>>(ei, nrm, hw, agg, E);
  bias_relu_kernel<8><<<g16, T, 0, stream>>>(agg, b3, hbuf, N * 16);

  // layer 4: 8 -> 4, dense [N,4] output
  gemm_wmma_kernel<8, 16><<<gG, 256, 0, stream>>>(hbuf, Wp4, hw, N);
  agg_init_kernel<<<g4, T, 0, stream>>>((const float4*)hw, dinv2, (float4*)agg, N * 4);
  scatter_kernel<4><<<gE, T, 0, stream>>>(ei, nrm, hw, agg, E);
  final_kernel<<<g4, T, 0, stream>>>(agg, b4, out, N * 4);
}